// MyModel_61933428416620
// MI455X (gfx1250) — compile-verified
//
#include <hip/hip_runtime.h>

// CTC forward loss (sum over batch), log-space, zero_infinity semantics.
// T=1000, B=32, C=1000, S=100 -> L=201 trellis columns.
//
// Strategy: 1 workgroup per batch element (32 WGs x 224 threads = 7 waves).
// alpha recursion in LDS (double buffered, one barrier/step). Emission
// gathers logp[t,b,ext[l]] are prefetched PF steps ahead with CDNA5 async
// global->LDS loads into a ring buffer; consumption gated by s_wait_asynccnt.
// Each wave reads only the ring columns its own lanes wrote, so the async
// pipeline needs no cross-wave synchronization.

#define NEGV  (-1e30f)
#define NT    224        // block size: 7 wave32s, covers L=201
#define PF    12         // async prefetch depth (outstanding per wave <= PF+1 <= 63)
#define NSLOT (PF + 1)   // ring slots (PF+1 avoids read/write slot collision)

#ifndef __has_builtin
#define __has_builtin(x) 0
#endif

#if __has_builtin(__builtin_amdgcn_global_load_async_to_lds_b32)
#define HAS_ASYNC_LDS 1
#else
#define HAS_ASYNC_LDS 0
#endif

#if __has_builtin(__builtin_amdgcn_s_wait_asynccnt)
#define HAS_WAIT_ASYNC 1
#else
#define HAS_WAIT_ASYNC 0
#endif

typedef __attribute__((address_space(1))) int as1_int;
typedef __attribute__((address_space(3))) int as3_int;

__device__ __forceinline__ void async_gather_b32(const float* g, float* l) {
#if HAS_ASYNC_LDS
  __builtin_amdgcn_global_load_async_to_lds_b32(
      (as1_int*)g, (as3_int*)l, 0, 0);
#else
  *l = *g;  // synchronous fallback (still correct, no latency hiding)
#endif
}

#if HAS_ASYNC_LDS
#if HAS_WAIT_ASYNC
#define WAIT_EMIT() do { __builtin_amdgcn_s_wait_asynccnt(PF); \
                         asm volatile("" ::: "memory"); } while (0)
#else
#define WAIT_EMIT() asm volatile("s_wait_asynccnt %0" :: "i"(PF) : "memory")
#endif
#else
#define WAIT_EMIT() asm volatile("" ::: "memory")
#endif

__device__ __forceinline__ float lse3f(float a, float b, float c) {
  float m = fmaxf(fmaxf(a, b), c);
  return m + __logf(__expf(a - m) + __expf(b - m) + __expf(c - m));
}

extern "C" __global__ __launch_bounds__(NT)
void ctc_scan_kernel(const float* __restrict__ logp,     // (T,B,C)
                     const int*   __restrict__ targets,  // (B,S)
                     const int*   __restrict__ in_lens,  // (B,)
                     const int*   __restrict__ tgt_lens, // (B,)
                     float*       __restrict__ loss_b,   // (B,)
                     int T, int B, int C, int S) {
  const int b  = blockIdx.x;
  const int l  = threadIdx.x;
  const int L  = 2 * S + 1;
  const int tl = tgt_lens[b];
  const int il = in_lens[b];
  const int Lb = 2 * tl + 1;

  __shared__ float s_alpha[2][NT + 2];      // +2 front pad (= NEG) for l-1, l-2
  __shared__ float s_ring[NSLOT][NT];       // async emission ring
  __shared__ float s_fin[2];                // end_blank, end_label

  // Extended label for this trellis column and skip-transition permission.
  int  ext_l = 0;      // blank for even l (and for idle lanes l >= L)
  bool skip  = false;
  if ((l & 1) && (l < L)) {
    const int* tb = targets + (size_t)b * S;
    ext_l = tb[l >> 1];
    if (l >= 3) skip = (ext_l != tb[(l >> 1) - 1]);
  }

  const size_t stride = (size_t)B * C;           // row stride per time step
  const float* ep = logp + (size_t)b * C + ext_l;

  // Init alpha buffers (incl. pads) to log-zero.
  s_alpha[0][l + 2] = NEGV;
  s_alpha[1][l + 2] = NEGV;
  if (l < 2) { s_alpha[0][l] = NEGV; s_alpha[1][l] = NEGV; }

  // Prime the async emission pipeline with steps 0..PF-1.
  for (int t = 0; t < PF; ++t) {
    int tp = (t < T) ? t : (T - 1);
    async_gather_b32(ep + (size_t)tp * stride, &s_ring[t % NSLOT][l]);
  }
  __syncthreads();

  const volatile float* ring = &s_ring[0][0];   // written by the async engine

  float a_self = NEGV;   // alpha[l] from previous step (register)
  float a_fin  = NEGV;   // snapshot at t == input_len-1
  int   p      = 0;      // read buffer
  for (int t = 0; t < T; ++t) {
    // Issue gather for step t+PF (clamped; re-reading the last row is benign).
    int tp = t + PF; if (tp >= T) tp = T - 1;
    async_gather_b32(ep + (size_t)tp * stride, &s_ring[(t + PF) % NSLOT][l]);

    WAIT_EMIT();  // oldest outstanding (step t) has landed in LDS
    float e = ring[(size_t)(t % NSLOT) * NT + l];

    float anew;
    if (t == 0) {
      anew = (l < 2) ? e : NEGV;                       // alpha0
    } else {
      float am1 = s_alpha[p][l + 1];                   // alpha[l-1]
      float am2 = skip ? s_alpha[p][l] : NEGV;         // alpha[l-2] if allowed
      anew = lse3f(a_self, am1, am2) + e;
    }
    if (l >= Lb) anew = NEGV;                          // valid mask
    a_self = anew;
    s_alpha[p ^ 1][l + 2] = anew;
    if (t == il - 1) a_fin = anew;
    __syncthreads();
    p ^= 1;
  }

  if (l == 2 * tl)               s_fin[0] = a_fin;     // end at blank
  if (tl > 0 && l == 2 * tl - 1) s_fin[1] = a_fin;     // end at last label
  __syncthreads();
  if (l == 0) {
    float eb  = s_fin[0];
    float elb = (tl > 0) ? s_fin[1] : NEGV;
    float m = fmaxf(eb, elb);
    float loss = -(m + __logf(__expf(eb - m) + __expf(elb - m)));
    if (loss > -0.5f * NEGV) loss = 0.0f;              // zero_infinity
    loss_b[b] = loss;
  }
}

extern "C" __global__ void ctc_reduce_kernel(const float* __restrict__ loss_b,
                                             float* __restrict__ out, int B) {
  if (threadIdx.x == 0 && blockIdx.x == 0) {
    float s = 0.0f;
    for (int i = 0; i < B; ++i) s += loss_b[i];   // fixed order: deterministic
    out[0] = s;
  }
}

extern "C" void kernel_launch(void* const* d_in, const int* in_sizes, int n_in,
                              void* d_out, int out_size, void* d_ws, size_t ws_size,
                              hipStream_t stream) {
  (void)n_in; (void)out_size; (void)ws_size;
  const float* logp     = (const float*)d_in[0];
  const int*   targets  = (const int*)d_in[1];
  const int*   in_lens  = (const int*)d_in[2];
  const int*   tgt_lens = (const int*)d_in[3];

  const int B = in_sizes[2];                       // from input_lens
  const int S = in_sizes[1] / B;                   // targets (B,S)
  const int C = 1000;                              // fixed by the reference model
  const int T = (int)((long long)in_sizes[0] / ((long long)B * (long long)C));

  float* loss_b = (float*)d_ws;                    // B floats of scratch

  ctc_scan_kernel<<<B, NT, 0, stream>>>(logp, targets, in_lens, tgt_lens,
                                        loss_b, T, B, C, S);
  ctc_reduce_kernel<<<1, 32, 0, stream>>>(loss_b, (float*)d_out, B);
}